// DECOMP_17626545783759
// MI455X (gfx1250) — compile-verified
//
#include <hip/hip_runtime.h>
#include <hip/hip_bf16.h>
#include <stdint.h>
#include <stddef.h>

// ---------------- problem constants ----------------
static constexpr int Bm    = 32;
static constexpr int Lm    = 4096;
static constexpr int Dm    = 512;
static constexpr int TCH   = 256;   // timesteps per chunk
static constexpr int KWARM = 128;   // warm-up steps; 0.7^128 ~ 1.6e-20 << fp32 eps
static constexpr int NS    = 8;     // async pipeline depth (power of 2)
static constexpr float ALPHA = 0.3f;

typedef float f32x2 __attribute__((ext_vector_type(2)));
typedef int   i32x2 __attribute__((ext_vector_type(2)));

#define ASG __attribute__((address_space(1)))
#define ASL __attribute__((address_space(3)))

// ---- CDNA5 async global->LDS copy (8 bytes per lane) ----
__device__ __forceinline__ void async_ld_b64(const f32x2* g, f32x2* l) {
#if __has_builtin(__builtin_amdgcn_global_load_async_to_lds_b64)
  __builtin_amdgcn_global_load_async_to_lds_b64(
      (ASG i32x2*)(void*)g, (ASL i32x2*)l, /*imm offset*/0, /*cpol*/0);
#else
  unsigned lo = (unsigned)(uintptr_t)(ASL void*)l;   // 32-bit LDS byte address
  asm volatile("global_load_async_to_lds_b64 %0, %1, off"
               :: "v"(lo), "v"(g) : "memory");
#endif
}

#define WAIT_ASYNC(n) asm volatile("s_wait_asynccnt %0" :: "i"(n) : "memory")
#define WAIT_DS0()    asm volatile("s_wait_dscnt 0" ::: "memory")

// One workgroup = one (batch b, time-chunk) pair, 256 threads x float2 = one
// contiguous 2KB row of D per timestep. 8-stage async double-buffer in LDS.
__global__ __launch_bounds__(256) void ema_chunk_kernel(
    const float* __restrict__ x, float* __restrict__ res, float* __restrict__ ma) {
  const int tid    = threadIdx.x;        // 0..255, owns d = 2*tid, 2*tid+1
  const int b      = blockIdx.x;         // 0..31
  const int chunk  = blockIdx.y;         // 0..15
  const int tstart = chunk * TCH;
  const int warm   = (chunk == 0) ? 0 : KWARM;
  const int t0     = tstart - warm;
  const int nT     = TCH + warm;

  __shared__ f32x2 stage[NS * 256];      // 16 KB: 8 row-slots of 2KB

  const f32x2* gx = (const f32x2*)(x + ((size_t)b * Lm + t0) * Dm) + tid;   // +256 per t
  f32x2* resp = (f32x2*)(res + ((size_t)b * Lm + tstart) * Dm) + tid;
  f32x2* map  = (f32x2*)(ma  + ((size_t)b * Lm + tstart) * Dm) + tid;

  // Prefill the pipeline: stages 0..NS-1 in flight.
#pragma unroll
  for (int i = 0; i < NS; ++i)
    async_ld_b64(gx + (size_t)i * 256, &stage[i * 256 + tid]);

  const float a = ALPHA, c = 1.0f - ALPHA;
  float s0 = 0.0f, s1 = 0.0f;

  auto compute_store = [&](int i, f32x2 xv) {
    if (i == 0) { s0 = xv.x; s1 = xv.y; }          // s_0 = x_0 (or warm-up seed)
    else        { s0 = a * xv.x + c * s0; s1 = a * xv.y + c * s1; }
    const int t = i - warm;                         // index within output chunk
    if (t >= 0) {
      f32x2 m = { s0, s1 };
      f32x2 r = { xv.x - s0, xv.y - s1 };
      __builtin_nontemporal_store(r, resp + (size_t)t * 256);
      __builtin_nontemporal_store(m, map  + (size_t)t * 256);
    }
  };

  // Steady state: consume stage i, refill same slot with stage i+NS.
  int i = 0;
  const int mainN = nT - NS;
  for (; i < mainN; ++i) {
    WAIT_ASYNC(NS - 1);                             // oldest async load landed
    f32x2 xv = stage[(i & (NS - 1)) * 256 + tid];   // ds_load_b64
    WAIT_DS0();                                     // slot free before async rewrite
    async_ld_b64(gx + (size_t)(i + NS) * 256, &stage[(i & (NS - 1)) * 256 + tid]);
    compute_store(i, xv);
  }

  // Drain the last NS stages.
  WAIT_ASYNC(0);
  for (; i < nT; ++i) {
    f32x2 xv = stage[(i & (NS - 1)) * 256 + tid];
    compute_store(i, xv);
  }
}

extern "C" void kernel_launch(void* const* d_in, const int* in_sizes, int n_in,
                              void* d_out, int out_size, void* d_ws, size_t ws_size,
                              hipStream_t stream) {
  const float* x = (const float*)d_in[0];
  float* res = (float*)d_out;                                   // output 0: x - ema
  float* ma  = (float*)d_out + (size_t)Bm * Lm * Dm;            // output 1: ema
  dim3 grid(Bm, Lm / TCH);                                      // 32 x 16 = 512 WGs
  ema_chunk_kernel<<<grid, 256, 0, stream>>>(x, res, ma);
}